// lateral_layer_9062380994837
// MI455X (gfx1250) — compile-verified
//
#include <hip/hip_runtime.h>
#include <stdint.h>

typedef float v2f __attribute__((ext_vector_type(2)));
typedef float v8f __attribute__((ext_vector_type(8)));

#define N_NEURONS 65536
#define N_GROUPS  1024              // N_NEURONS / 64
#define BATCH     1024
#define TB        32                // batch rows per wave (pipeline depth over this loop)
#define WAVES_PER_BLOCK 8
#define NUM_BLOCKS ((N_GROUPS * (BATCH / TB)) / WAVES_PER_BLOCK)  // 4096

// Cross-lane xor-add within aligned 16-lane rows via DPP ROW_XMASK (0x160|mask).
// Pure VALU: no LDS pipe traffic, no s_wait_dscnt, unlike ds_bpermute-based __shfl_xor.
#define DPP_ROW_XOR_ADD(v, mask)                                              \
    (v) += __int_as_float(__builtin_amdgcn_update_dpp(                        \
        0, __float_as_int(v), 0x160 | (mask), 0xf, 0xf, true))

__global__ __launch_bounds__(256) void lateral_norm_kernel(
    const float* __restrict__ X, const int* __restrict__ IDX, float* __restrict__ OUT)
{
    // 8 waves * 2 stages * 64 floats = 4 KB of LDS per block (wave-private slices)
    __shared__ float smem[WAVES_PER_BLOCK * 2 * 64];

    const int lane = threadIdx.x & 31;
    const int warp = threadIdx.x >> 5;
    const int wave = blockIdx.x * WAVES_PER_BLOCK + warp;
    const int g    = wave & (N_GROUPS - 1);   // group id
    const int b0   = (wave >> 10) * TB;       // first batch row of this wave's tile

    // Each lane owns 2 neurons of the group: j = g*64 + 2*lane (+1).
    const int2 ip  = *reinterpret_cast<const int2*>(IDX + g * 64 + 2 * lane);
    const long long off0 = (long long)ip.x;   // original neuron positions (load==store addr)
    const long long off1 = (long long)ip.y;

    float* wbuf = &smem[warp * 128];                       // 2 stages x 64 floats
    const uint32_t wbase = (uint32_t)(uintptr_t)wbuf;      // LDS byte offset (low 32 bits)
    const uint32_t lslot = wbase + (uint32_t)lane * 8u;    // this lane's 8-byte slot, stage 0

    // ---- prime the async pipeline: gather rows b0 (stage 0) and b0+1 (stage 1) into LDS ----
    {
        const float* r0 = X + (size_t)b0 * N_NEURONS;
        uint64_t ga = (uint64_t)(uintptr_t)(r0 + off0);
        uint64_t gb = (uint64_t)(uintptr_t)(r0 + off1);
        asm volatile("global_load_async_to_lds_b32 %0, %1, off" :: "v"(lslot),      "v"(ga) : "memory");
        asm volatile("global_load_async_to_lds_b32 %0, %1, off" :: "v"(lslot + 4u), "v"(gb) : "memory");

        const float* r1 = X + (size_t)(b0 + 1) * N_NEURONS;
        uint64_t gc = (uint64_t)(uintptr_t)(r1 + off0);
        uint64_t gd = (uint64_t)(uintptr_t)(r1 + off1);
        asm volatile("global_load_async_to_lds_b32 %0, %1, off" :: "v"(lslot + 256u), "v"(gc) : "memory");
        asm volatile("global_load_async_to_lds_b32 %0, %1, off" :: "v"(lslot + 260u), "v"(gd) : "memory");
    }

    for (int i = 0; i < TB; ++i) {
        const int st = i & 1;

        // Wait for the current stage's 2 async loads (async loads complete in order).
        if (i == TB - 1) { asm volatile("s_wait_asynccnt 0x0" ::: "memory"); }
        else             { asm volatile("s_wait_asynccnt 0x2" ::: "memory"); }

        // Read this lane's two gathered activations back from LDS.
        float2 e = *reinterpret_cast<float2*>(wbuf + st * 64 + lane * 2);

        // Group sum via V_WMMA_F32_16X16X4_F32 (exact f32 adds):
        //   A = ones(16x4), B = 64 gathered values (2 VGPRs) -> D[m,n] = sum_k B[k,n].
        // One WMMA reduces 64 -> 16 column partials (replicated over both lane halves).
        v2f a;  a[0] = 1.0f; a[1] = 1.0f;
        v2f bm; bm[0] = e.x; bm[1] = e.y;
        v8f c = {0.f, 0.f, 0.f, 0.f, 0.f, 0.f, 0.f, 0.f};
        v8f d = __builtin_amdgcn_wmma_f32_16x16x4_f32(
            /*neg_a=*/false, a, /*neg_b=*/false, bm,
            /*c_mod=*/(short)0, c, /*reuse_a=*/false, /*reuse_b=*/false);

        // Finish the 16-way lane reduction with DPP ROW_XMASK adds (VALU-only).
        float t = d[0];                       // t[n] at lane n and lane n+16
        DPP_ROW_XOR_ADD(t, 1);
        DPP_ROW_XOR_ADD(t, 2);
        DPP_ROW_XOR_ADD(t, 4);
        DPP_ROW_XOR_ADD(t, 8);                // every lane now holds the full group sum

        const float r = __builtin_amdgcn_rcpf(t);   // ~1 ulp; avoids full div sequence

        float* orow = OUT + (size_t)(b0 + i) * N_NEURONS;
        orow[off0] = e.x * r;
        orow[off1] = e.y * r;

        // Refill this stage buffer with row b0+i+2 (after ensuring our ds reads drained).
        if (i + 2 < TB) {
            asm volatile("s_wait_dscnt 0x0" ::: "memory");   // WAR guard on LDS stage reuse
            const float* rn = X + (size_t)(b0 + i + 2) * N_NEURONS;
            uint64_t ga = (uint64_t)(uintptr_t)(rn + off0);
            uint64_t gb = (uint64_t)(uintptr_t)(rn + off1);
            const uint32_t la = lslot + (uint32_t)st * 256u;
            asm volatile("global_load_async_to_lds_b32 %0, %1, off" :: "v"(la),      "v"(ga) : "memory");
            asm volatile("global_load_async_to_lds_b32 %0, %1, off" :: "v"(la + 4u), "v"(gb) : "memory");
        }
    }
}

extern "C" void kernel_launch(void* const* d_in, const int* in_sizes, int n_in,
                              void* d_out, int out_size, void* d_ws, size_t ws_size,
                              hipStream_t stream) {
    const float* x   = (const float*)d_in[0];
    const int*   idx = (const int*)d_in[1];
    float*       out = (float*)d_out;
    (void)in_sizes; (void)n_in; (void)out_size; (void)d_ws; (void)ws_size;

    hipLaunchKernelGGL(lateral_norm_kernel,
                       dim3(NUM_BLOCKS), dim3(256), 0, stream,
                       x, idx, out);
}